// UnaryEdgeDetect_24515673326111
// MI455X (gfx1250) — compile-verified
//
#include <hip/hip_runtime.h>
#include <stdint.h>

// Problem geometry (from reference): T=256, H=W=255
#define T_STEPS 256
#define HW      (255 * 255)   // 65025 pixels
#define BLOCK   256
#define PIPE    16            // async pipeline depth (stages of t prefetched ahead)

typedef __attribute__((address_space(1))) int GInt;  // global int
typedef __attribute__((address_space(3))) int LInt;  // LDS int

// One 4-byte async copy global -> LDS (tracked by ASYNCcnt, no VGPR data path).
__device__ __forceinline__ void async_copy_b32(const float* g, float* l) {
#if __has_builtin(__builtin_amdgcn_global_load_async_to_lds_b32)
  __builtin_amdgcn_global_load_async_to_lds_b32((GInt*)g, (LInt*)l, 0, 0);
#else
  unsigned lofs = (unsigned)(uintptr_t)(LInt*)l;
  asm volatile("global_load_async_to_lds_b32 %0, %1, off"
               :: "v"(lofs), "v"(g) : "memory");
#endif
}

// Raw waits with memory clobber so the compiler cannot move LDS reads across them.
#define WAIT_ASYNCCNT(n) asm volatile("s_wait_asynccnt %0" :: "i"(n) : "memory")
#define WAIT_DSCNT0()    asm volatile("s_wait_dscnt 0" ::: "memory")

__global__ void __launch_bounds__(BLOCK)
unary_edge_detect_kernel(const float* __restrict__ in_ij,    // Pr(i,j)
                         const float* __restrict__ in_i1j1,  // Pr(i+1,j+1)
                         const float* __restrict__ in_i1j,   // Pr(i+1,j)
                         const float* __restrict__ in_ij1,   // Pr(i,j+1)
                         float* __restrict__ out) {
  // PIPE-deep staging: 2 operands per t-slice per pixel.
  // Even t: A=Pr(i,j),     B=Pr(i,j+1)     (sel=0 picks these in the MUXes)
  // Odd  t: A=Pr(i+1,j+1), B=Pr(i+1,j)     (sel=1 picks 1-these)
  __shared__ float bufA[PIPE][BLOCK];
  __shared__ float bufB[PIPE][BLOCK];

  const int tid = threadIdx.x;
  int gp = blockIdx.x * BLOCK + tid;
  if (gp >= HW) gp = HW - 1;  // clamp tail: duplicate lanes store identical values

  // Prologue: prefetch stages 0..PIPE-1 (2 async ops per stage per lane).
  #pragma unroll
  for (int t = 0; t < PIPE; ++t) {
    const float* a = (t & 1) ? in_i1j1 : in_ij;
    const float* b = (t & 1) ? in_i1j  : in_ij1;
    const size_t off = (size_t)t * HW + (size_t)gp;
    async_copy_b32(a + off, &bufA[t][tid]);
    async_copy_b32(b + off, &bufB[t][tid]);
  }

  // UnaryAbs saturating up/down counters, init at midpoint 128.
  int accx = 128;
  int accy = 128;

  for (int t = 0; t < T_STEPS; ++t) {
    // Allow the (PIPE-1) future stages (2 ops each) to stay in flight;
    // guarantees stage t has landed in LDS.
    WAIT_ASYNCCNT(2 * (PIPE - 1));

    const int slot = t & (PIPE - 1);
    const int s = t & 1;                       // Sobol sel bit: t % 2
    const int ia = bufA[slot][tid] > 0.5f;     // bit in {0,1}
    const int ib = bufB[slot][tid] > 0.5f;

    // Gaines scaled add via MUX: g = sel ? (1 - bit) : bit  ==  bit ^ sel
    const int gx = ia ^ s;
    const int gy = ib ^ s;

    // Saturating counters: acc = clip(acc + (2g - 1), 0, 255)
    accx += 2 * gx - 1;
    accx = accx < 0 ? 0 : (accx > 255 ? 255 : accx);
    accy += 2 * gy - 1;
    accy = accy < 0 ? 0 : (accy > 255 ? 255 : accy);

    // Bipolar abs: flip bit when counter below midpoint.
    const int gxa = gx ^ (accx < 128);
    const int gya = gy ^ (accy < 128);

    // Final Gaines add of |Gx|,|Gy|: MUX by sel.
    const float ps = (float)(s ? gya : gxa);
    __builtin_nontemporal_store(ps, out + (size_t)t * HW + (size_t)gp);

    // Ensure our ds_loads of this slot completed before the async engine
    // overwrites it (async LDS writes are unordered w.r.t. DS ops).
    WAIT_DSCNT0();

    const int tn = t + PIPE;
    if (tn < T_STEPS) {
      const float* a = (tn & 1) ? in_i1j1 : in_ij;
      const float* b = (tn & 1) ? in_i1j  : in_ij1;
      const size_t off = (size_t)tn * HW + (size_t)gp;
      async_copy_b32(a + off, &bufA[slot][tid]);
      async_copy_b32(b + off, &bufB[slot][tid]);
    }
  }
}

extern "C" void kernel_launch(void* const* d_in, const int* in_sizes, int n_in,
                              void* d_out, int out_size, void* d_ws, size_t ws_size,
                              hipStream_t stream) {
  // setup_inputs() order: inp_Pr_i_j, inp_Pr_i1_j1, inp_Pr_i1_j, inp_Pr_i_j1
  const float* in_ij   = (const float*)d_in[0];
  const float* in_i1j1 = (const float*)d_in[1];
  const float* in_i1j  = (const float*)d_in[2];
  const float* in_ij1  = (const float*)d_in[3];
  float* out = (float*)d_out;

  dim3 grid((HW + BLOCK - 1) / BLOCK);  // 255 blocks
  dim3 block(BLOCK);
  unary_edge_detect_kernel<<<grid, block, 0, stream>>>(in_ij, in_i1j1, in_i1j,
                                                       in_ij1, out);
}